// CEGALayer_63694365000002
// MI455X (gfx1250) — compile-verified
//
#include <hip/hip_runtime.h>
#include <hip/hip_bf16.h>
#include <math.h>

// ---------------- Clifford algebra tables (Cl(4,1), basis sorted by (grade, mask)) ----
__constant__ int c_MASK[32] = {
    0,
    1,2,4,8,16,
    3,5,6,9,10,12,17,18,20,24,
    7,11,13,14,19,21,22,25,26,28,
    15,23,27,29,30,
    31};
__constant__ int c_IDX[32] = { // mask -> index
    0,1,2,6,3,7,8,16,4,9,10,17,11,18,19,26,
    5,12,13,20,14,21,22,27,15,23,24,28,25,29,30,31};

// signed coefficient of e_ma * e_mb (metric +,+,+,+,- ; bit4 = e5)
__device__ __forceinline__ float blade_sign(int ma, int mb) {
    int s = 0;
    int x = ma >> 1;
    while (x) { s += __popc(x & mb); x >>= 1; }
    float sgn = (s & 1) ? -1.f : 1.f;
    if ((ma & mb) & 16) sgn = -sgn;   // e5*e5 = -1
    return sgn;
}

// ---------------- fp32 WMMA GEMM: C[M,N] = A[M,K] @ B[K,N] (+bias) -------------------
typedef float v2f __attribute__((ext_vector_type(2)));
typedef float v8f __attribute__((ext_vector_type(8)));

__global__ void k_gemm_wmma(const float* __restrict__ A, const float* __restrict__ B,
                            const float* __restrict__ bias, float* __restrict__ C,
                            int K, int lda, int ldb, int ldc) {
    const int mt   = blockIdx.x * 16;
    const int nt   = blockIdx.y * 16;
    const int lane = threadIdx.x;      // 32 threads = 1 wave
    const int half = lane >> 4;        // 0: K pair {0,1}, 1: K pair {2,3}
    const int lo   = lane & 15;

    v8f acc = {};
    for (int kb = 0; kb < K; kb += 4) {
        const int ka = kb + 2 * half;
        v2f a, b;
        a.x = A[(mt + lo) * lda + ka];
        a.y = A[(mt + lo) * lda + ka + 1];
        b.x = B[ka * ldb + nt + lo];
        b.y = B[(ka + 1) * ldb + nt + lo];
        acc = __builtin_amdgcn_wmma_f32_16x16x4_f32(false, a, false, b,
                                                    (short)0, acc, false, false);
    }
    const int col = nt + lo;
    const float bv = bias ? bias[col] : 0.f;
    #pragma unroll
    for (int r = 0; r < 8; ++r) {
        const int row = mt + r + 8 * half;
        C[row * ldc + col] = acc[r] + bv;
    }
}

// ---------------- prep kernels -------------------------------------------------------
// h_flat[n][c*32+k] = h[n][k][c]
__global__ void k_hflat(const float* __restrict__ H, float* __restrict__ hflat) {
    const int n = blockIdx.x, t = threadIdx.x;          // 512
    hflat[n * 512 + t] = H[(n * 32 + (t & 31)) * 16 + (t >> 5)];
}

// invariants (96 x 48) and conformal embedding p5 (96 x 5)
__global__ void k_inv(const float* __restrict__ H, const float* __restrict__ coords,
                      float* __restrict__ inv, float* __restrict__ p5) {
    const int n = blockIdx.x, c = threadIdx.x;          // 16
    inv[n * 48 + c] = H[(n * 32 + 0) * 16 + c];
    float v2 = 0.f;
    for (int k = 1; k <= 3; ++k) { float x = H[(n * 32 + k) * 16 + c]; v2 += x * x; }
    inv[n * 48 + 16 + c] = sqrtf(v2);
    float b2 = 0.f;
    for (int k = 6; k <= 15; ++k) { float x = H[(n * 32 + k) * 16 + c]; b2 += x * x; }
    inv[n * 48 + 32 + c] = sqrtf(b2);
    if (c == 0) {
        float x = coords[n * 3 + 0], y = coords[n * 3 + 1], z = coords[n * 3 + 2];
        float sq = x * x + y * y + z * z;
        p5[n * 5 + 0] = x; p5[n * 5 + 1] = y; p5[n * 5 + 2] = z;
        p5[n * 5 + 3] = 0.5f * sq - 0.5f;
        p5[n * 5 + 4] = 0.5f * sq + 0.5f;
    }
}

// pad Wm2 (128x24)->(128x32), bm2 (24)->(32)
__global__ void k_pad(const float* __restrict__ Wm2, const float* __restrict__ bm2,
                      float* __restrict__ Wm2p, float* __restrict__ bm2p) {
    const int t = threadIdx.x;                          // 256
    for (int idx = t; idx < 128 * 32; idx += 256) {
        const int r = idx >> 5, col = idx & 31;
        Wm2p[idx] = (col < 24) ? Wm2[r * 24 + col] : 0.f;
    }
    if (t < 32) bm2p[t] = (t < 24) ? bm2[t] : 0.f;
}

// hidden[e][t] = relu(Ai[i][t] + Bj[j][t] + bm1[t])
__global__ void k_hidden(const float* __restrict__ Ai, const float* __restrict__ Bj,
                         const float* __restrict__ bm1, float* __restrict__ hidden) {
    const int e = blockIdx.x, t = threadIdx.x;          // 9216 x 128
    const int i = e / 96, j = e % 96;
    const float x = Ai[i * 128 + t] + Bj[j * 128 + t] + bm1[t];
    hidden[e * 128 + t] = x > 0.f ? x : 0.f;
}

// ---------------- attention (conformal dot + mask + softmax) -------------------------
__global__ void k_attn(const float* __restrict__ p5, const int* __restrict__ adj,
                       float* __restrict__ attn) {
    const int i = blockIdx.x, j = threadIdx.x;          // 128 threads, 96 used
    __shared__ float lrow[96];
    __shared__ float mx_s, sum_s;
    if (j < 96) {
        float l = 0.f;
        for (int m = 0; m < 4; ++m) l += p5[i * 5 + m] * p5[j * 5 + m];
        l -= p5[i * 5 + 4] * p5[j * 5 + 4];
        if (adj[i * 96 + j] == 0) l = -1e9f;
        lrow[j] = l;
    }
    __syncthreads();
    if (j == 0) {
        float mx = -3.0e38f;
        for (int t = 0; t < 96; ++t) mx = fmaxf(mx, lrow[t]);
        float sm = 0.f;
        for (int t = 0; t < 96; ++t) sm += expf(lrow[t] - mx);
        mx_s = mx; sum_s = sm;
    }
    __syncthreads();
    if (j < 96) attn[i * 96 + j] = expf(lrow[j] - mx_s) / sum_s;
}

// ---------------- motors: gp(translator, rotor), sparse 7x4 blade products -----------
__global__ void k_motor(const float* __restrict__ mp, float* __restrict__ motors) {
    const int t = blockIdx.x * blockDim.x + threadIdx.x; // e*4+h
    if (t >= 9216 * 4) return;
    const int e = t >> 2, h = t & 3;
    const float* p = mp + e * 32 + h * 6;
    const float b0 = p[0], b1 = p[1], b2 = p[2];
    const float bn = sqrtf(b0 * b0 + b1 * b1 + b2 * b2 + 1e-8f);
    const float u0 = b0 / bn, u1 = b1 / bn, u2 = b2 / bn;
    const float cc = cosf(0.5f * bn), ss = sinf(0.5f * bn);
    float R[4]; const int RI[4] = {0, 6, 7, 8};            // scalar, e12, e13, e23
    R[0] = cc; R[1] = ss * u2; R[2] = -ss * u1; R[3] = ss * u0;
    float T[7]; const int TI[7] = {0, 9, 10, 11, 12, 13, 14};
    T[0] = 1.f;
    T[1] = -0.5f * p[3]; T[2] = -0.5f * p[4]; T[3] = -0.5f * p[5];
    T[4] = -0.5f * p[3]; T[5] = -0.5f * p[4]; T[6] = -0.5f * p[5];
    float m[32];
    #pragma unroll
    for (int k = 0; k < 32; ++k) m[k] = 0.f;
    for (int a = 0; a < 7; ++a) {
        const int ma = c_MASK[TI[a]];
        for (int b = 0; b < 4; ++b) {
            const int mb = c_MASK[RI[b]];
            m[c_IDX[ma ^ mb]] += blade_sign(ma, mb) * T[a] * R[b];
        }
    }
    float* o = motors + t * 32;
    #pragma unroll
    for (int k = 0; k < 32; ++k) o[k] = m[k];
}

// ---------------- edge sandwich m * v * ~m, attention-weighted sum over j ------------
// block = (i, h), 512 threads: thread -> (c = tid>>5, k = tid&31)
__global__ void k_agg(const float* __restrict__ motors, const float* __restrict__ V,
                      const float* __restrict__ attn, float* __restrict__ aggh) {
    const int i = blockIdx.x, h = blockIdx.y;
    const int tid = threadIdx.x;
    const int c = tid >> 5, k = tid & 31;

    __shared__ float SGN[32][32];
    __shared__ int   sM[32], sI[32];
    __shared__ float m_sh[32], mr_sh[32];
    __shared__ float v_sh[16][32], t_sh[16][32];
    __shared__ float a_sh;

    if (tid < 32) {
        sM[tid] = c_MASK[tid];
        sI[tid] = c_IDX[tid];
    }
    for (int idx = tid; idx < 1024; idx += 512) {
        const int ia = idx >> 5, ib = idx & 31;
        SGN[ia][ib] = blade_sign(c_MASK[ia], c_MASK[ib]);
    }
    __syncthreads();
    const int mk = sM[k];

    float acc = 0.f;
    for (int j = 0; j < 96; ++j) {
        __syncthreads();
        if (tid < 32) {
            const float mv = motors[((i * 96 + j) * 4 + h) * 32 + tid];
            m_sh[tid] = mv;
            const int g = __popc(sM[tid]);
            mr_sh[tid] = (((g * (g - 1)) >> 1) & 1) ? -mv : mv;  // reversion sign
        }
        v_sh[c][k] = V[j * 2048 + h * 512 + c * 32 + k];
        if (tid == 0) a_sh = attn[i * 96 + j];
        __syncthreads();

        // t = gp(v_c, reverse(m)) : t_k = sum_ib sign(ib,pb) v[ib] mr[pb], mask_pb = mask_ib ^ mask_k
        float tacc = 0.f;
        #pragma unroll
        for (int ib = 0; ib < 32; ++ib) {
            const int pb = sI[sM[ib] ^ mk];
            tacc += SGN[ib][pb] * v_sh[c][ib] * mr_sh[pb];
        }
        t_sh[c][k] = tacc;
        __syncthreads();

        // out = gp(m, t) : out_k = sum_ia sign(ia,pa) m[ia] t[pa], mask_pa = mask_ia ^ mask_k
        float oacc = 0.f;
        #pragma unroll
        for (int ia = 0; ia < 32; ++ia) {
            const int pa = sI[sM[ia] ^ mk];
            oacc += SGN[ia][pa] * m_sh[ia] * t_sh[c][pa];
        }
        acc += a_sh * oacc;
    }
    aggh[((i * 4 + h) * 32 + k) * 16 + c] = acc;
}

// ---------------- residual + head-mean + LayerNorm -----------------------------------
__global__ void k_finalize(const float* __restrict__ hres, const float* __restrict__ aggh,
                           const float* __restrict__ lnw, const float* __restrict__ lnb,
                           float* __restrict__ out) {
    const int n = blockIdx.x, tid = threadIdx.x;        // 512
    const int k = tid >> 4, c = tid & 15;
    float val = hres[n * 512 + c * 32 + k];             // (N, Cout, 32) -> transposed read
    float s = 0.f;
    for (int h = 0; h < 4; ++h) s += aggh[((n * 4 + h) * 32 + k) * 16 + c];
    val += 0.25f * s;

    __shared__ float red[512];
    __shared__ float mu_s, var_s;
    red[tid] = val; __syncthreads();
    for (int st = 256; st > 0; st >>= 1) { if (tid < st) red[tid] += red[tid + st]; __syncthreads(); }
    if (tid == 0) mu_s = red[0] / 512.f;
    __syncthreads();
    const float d = val - mu_s;
    red[tid] = d * d; __syncthreads();
    for (int st = 256; st > 0; st >>= 1) { if (tid < st) red[tid] += red[tid + st]; __syncthreads(); }
    if (tid == 0) var_s = red[0] / 512.f;
    __syncthreads();
    const float r = rsqrtf(var_s + 1e-5f);
    out[(n * 32 + k) * 16 + c] = d * r * lnw[k * 16 + c] + lnb[k * 16 + c];
}

// ---------------- driver -------------------------------------------------------------
extern "C" void kernel_launch(void* const* d_in, const int* in_sizes, int n_in,
                              void* d_out, int out_size, void* d_ws, size_t ws_size,
                              hipStream_t stream) {
    const float* H      = (const float*)d_in[0];   // (1,96,32,16)
    const float* coords = (const float*)d_in[1];   // (1,96,3)
    const int*   adj    = (const int*)  d_in[2];   // (1,96,96)
    const float* Wm1    = (const float*)d_in[3];   // (96,128)
    const float* bm1    = (const float*)d_in[4];   // (128)
    const float* Wm2    = (const float*)d_in[5];   // (128,24)
    const float* bm2    = (const float*)d_in[6];   // (24)
    const float* Wv     = (const float*)d_in[7];   // (512,2048)
    const float* bv     = (const float*)d_in[8];   // (2048)
    const float* Wu     = (const float*)d_in[9];   // (512,512)
    const float* bu     = (const float*)d_in[10];  // (512)
    const float* lnw    = (const float*)d_in[11];  // (32,16)
    const float* lnb    = (const float*)d_in[12];  // (32,16)
    float* out = (float*)d_out;

    float* w = (float*)d_ws;
    float* inv    = w; w += 96 * 48;
    float* p5     = w; w += 96 * 5;
    float* attn   = w; w += 96 * 96;
    float* hflat  = w; w += 96 * 512;
    float* Ai     = w; w += 96 * 128;
    float* Bj     = w; w += 96 * 128;
    float* hidden = w; w += 9216 * 128;
    float* Wm2p   = w; w += 128 * 32;
    float* bm2p   = w; w += 32;
    float* mp     = w; w += 9216 * 32;
    float* motors = w; w += 9216 * 4 * 32;
    float* V      = w; w += 96 * 2048;
    float* hres   = w; w += 96 * 512;
    float* aggh   = w; w += 96 * 4 * 32 * 16;

    // prep
    k_hflat<<<96, 512, 0, stream>>>(H, hflat);
    k_inv  <<<96, 16,  0, stream>>>(H, coords, inv, p5);
    k_pad  <<<1, 256,  0, stream>>>(Wm2, bm2, Wm2p, bm2p);
    k_attn <<<96, 128, 0, stream>>>(p5, adj, attn);

    // pair-MLP, factored: Ai = inv @ Wm1[:48], Bj = inv @ Wm1[48:]
    k_gemm_wmma<<<dim3(6, 8),   32, 0, stream>>>(inv, Wm1,            nullptr, Ai, 48, 48, 128, 128);
    k_gemm_wmma<<<dim3(6, 8),   32, 0, stream>>>(inv, Wm1 + 48 * 128, nullptr, Bj, 48, 48, 128, 128);
    k_hidden<<<9216, 128, 0, stream>>>(Ai, Bj, bm1, hidden);
    k_gemm_wmma<<<dim3(576, 2), 32, 0, stream>>>(hidden, Wm2p, bm2p, mp, 128, 128, 32, 32);

    // motors from params
    k_motor<<<144, 256, 0, stream>>>(mp, motors);

    // value / update projections
    k_gemm_wmma<<<dim3(6, 128), 32, 0, stream>>>(hflat, Wv, bv, V,    512, 512, 2048, 2048);
    k_gemm_wmma<<<dim3(6, 32),  32, 0, stream>>>(hflat, Wu, bu, hres, 512, 512, 512,  512);

    // sandwich + attention aggregation, then residual + LN
    k_agg<<<dim3(96, 4), 512, 0, stream>>>(motors, V, attn, aggh);
    k_finalize<<<96, 512, 0, stream>>>(hres, aggh, lnw, lnb, out);

    (void)in_sizes; (void)n_in; (void)out_size; (void)ws_size;
}